// ParallelOuterProductMean_30090540875871
// MI455X (gfx1250) — compile-verified
//
#include <hip/hip_runtime.h>
#include <hip/hip_bf16.h>

typedef __attribute__((ext_vector_type(2))) float v2f;
typedef __attribute__((ext_vector_type(8))) float v8f;

#define S_DIM 128
#define N_DIM 384
#define CM    256
#define CH    32
#define CZ    128
#define TI    4
#define TJ    4

// ---------------------------------------------------------------------------
// CDNA5 async global->LDS copy (ASYNCcnt-tracked, bypasses VGPRs).
// Generic pointers into LDS have their LDS byte offset in the low 32 bits
// (ISA 10.2: for the LDS aperture the low 32 bits give the LDS address).
// ---------------------------------------------------------------------------
__device__ __forceinline__ void async_ld_b128(void* lds_dst, const void* gsrc) {
    unsigned lds_addr = (unsigned)(unsigned long long)lds_dst;
    asm volatile("global_load_async_to_lds_b128 %0, %1, off"
                 :: "v"(lds_addr), "v"(gsrc)
                 : "memory");
}

__device__ __forceinline__ void wait_async0() {
#if __has_builtin(__builtin_amdgcn_s_wait_asynccnt)
    __builtin_amdgcn_s_wait_asynccnt(0);
#else
    asm volatile("s_wait_asynccnt 0" ::: "memory");
#endif
}

// ---------------------------------------------------------------------------
// Kernel 1: LayerNorm over C_m, then a = ln@w1+b1, b = ln@w2+b2, * mask.
// Writes A,B in transposed layout [N, S, CH] so the fused kernel reads
// contiguous (i, s, c) panels. One block of 256 threads per (s,n) row.
// ---------------------------------------------------------------------------
__global__ void opm_ln_proj_kernel(const float* __restrict__ m,
                                   const float* __restrict__ mask,
                                   const float* __restrict__ gamma,
                                   const float* __restrict__ beta,
                                   const float* __restrict__ w1,
                                   const float* __restrict__ b1,
                                   const float* __restrict__ w2,
                                   const float* __restrict__ b2,
                                   float* __restrict__ Ag,
                                   float* __restrict__ Bg)
{
    const int r = blockIdx.x;           // 0 .. S*N-1
    const int s = r / N_DIM;
    const int n = r - s * N_DIM;
    const int t = threadIdx.x;          // 0 .. 255

    __shared__ float xs[CM];
    __shared__ float red[CM];
    __shared__ float parts[256];        // 4 k-segments x 64 outputs

    const float x = m[(size_t)r * CM + t];

    // mean
    red[t] = x;
    __syncthreads();
    for (int off = 128; off > 0; off >>= 1) {
        if (t < off) red[t] += red[t + off];
        __syncthreads();
    }
    const float mu = red[0] * (1.0f / CM);
    __syncthreads();

    // variance (E[(x-mu)^2])
    const float xc = x - mu;
    red[t] = xc * xc;
    __syncthreads();
    for (int off = 128; off > 0; off >>= 1) {
        if (t < off) red[t] += red[t + off];
        __syncthreads();
    }
    const float var = red[0] * (1.0f / CM);
    __syncthreads();

    const float ln = xc * rsqrtf(var + 1e-5f) * gamma[t] + beta[t];
    xs[t] = ln;
    __syncthreads();

    // 64 outputs (32 for a via w1, 32 for b via w2); 4-way split-K over CM.
    const int c    = t & 63;
    const int kseg = t >> 6;            // 0..3
    const int cc   = c & 31;
    const float* W = (c < 32) ? w1 : w2;
    float p = 0.0f;
    const int k0 = kseg * 64;
    #pragma unroll 8
    for (int k = k0; k < k0 + 64; ++k)
        p += xs[k] * W[k * CH + cc];
    parts[kseg * 64 + c] = p;
    __syncthreads();

    if (t < 64) {
        float v = parts[t] + parts[64 + t] + parts[128 + t] + parts[192 + t];
        const float mk = mask[r];
        const int ch = t & 31;
        if (t < 32) {
            v = (v + b1[ch]) * mk;
            Ag[((size_t)n * S_DIM + s) * CH + ch] = v;
        } else {
            v = (v + b2[ch]) * mk;
            Bg[((size_t)n * S_DIM + s) * CH + ch] = v;
        }
    }
}

// ---------------------------------------------------------------------------
// Kernel 2: fused outer-product + projection, all-WMMA (V_WMMA_F32_16X16X4_F32).
//   Stage A: G[m=i*32+c][n2=j*32+d] = sum_s A[i][s][c] * B[j][s][d]
//            (128x128 GEMM, K=128) accumulated via wmma, result to LDS.
//   Stage B: out[p=i*4+j][z] = sum_q G-view[p][q] * w_out[q][z]
//            (16x128 GEMM, K=1024) straight from the LDS G tile.
// Block: 256 threads (8 waves). Dynamic LDS: As 64KB + Bs 64KB + G 64KB.
// Panels arrive via CDNA5 async global->LDS DMA (ASYNCcnt).
// ---------------------------------------------------------------------------
__global__ void opm_fused_wmma_kernel(const float* __restrict__ Ag,
                                      const float* __restrict__ Bg,
                                      const float* __restrict__ mask,
                                      const float* __restrict__ w_out,
                                      const float* __restrict__ b_out,
                                      float* __restrict__ out)
{
    extern __shared__ float lds[];
    float* As = lds;                    // [TI][S][CH]  -> (i*S + s)*CH + c
    float* Bs = lds + TI * S_DIM * CH;  // [TJ][S][CH]
    float* G  = lds + 2 * TI * S_DIM * CH;  // [128][128]
    __shared__ float normS[TI * TJ];

    const int iBase = blockIdx.y * TI;
    const int jBase = blockIdx.x * TJ;
    const int t    = threadIdx.x;       // 0..255
    const int lane = t & 31;
    const int wave = t >> 5;            // 0..7
    const int half = lane >> 4;         // 0/1 (K-pair select per ISA layout)
    const int l15  = lane & 15;

    // ---- async DMA of the A/B panels straight into LDS ----
    const float4* aSrc = (const float4*)(Ag + (size_t)iBase * S_DIM * CH);
    const float4* bSrc = (const float4*)(Bg + (size_t)jBase * S_DIM * CH);
    float4* As4 = (float4*)As;
    float4* Bs4 = (float4*)Bs;
    const int nVec = TI * S_DIM * CH / 4;   // 4096 float4 per panel
    for (int k = t; k < nVec; k += 256) {
        async_ld_b128(As4 + k, aSrc + k);
        async_ld_b128(Bs4 + k, bSrc + k);
    }

    // ---- norm tile = (mask^T mask)[i,j] + eps (overlaps the DMA) ----
    if (t < TI * TJ) {
        const int i = iBase + (t >> 2);
        const int j = jBase + (t & 3);
        float acc = 0.0f;
        for (int s2 = 0; s2 < S_DIM; ++s2)
            acc += mask[s2 * N_DIM + i] * mask[s2 * N_DIM + j];
        normS[t] = acc + 1e-3f;
    }

    wait_async0();
    __syncthreads();

    // ---- Stage A: 64 16x16 tiles of the 128x128 Gram matrix; 8 per wave ----
    for (int tt = wave * 8; tt < wave * 8 + 8; ++tt) {
        const int mT = (tt >> 3) * 16;
        const int nT = (tt & 7) * 16;
        const int mRow = mT + l15;      // A-matrix: M = lane&15 (both halves)
        const int ai = mRow >> 5, ac = mRow & 31;
        const int nCol = nT + l15;      // B-matrix: N = lane&15 (both halves)
        const int bj = nCol >> 5, bd = nCol & 31;

        v8f acc = {};
        for (int k = 0; k < S_DIM; k += 4) {
            const int ka = k + 2 * half;
            v2f af, bf;
            af.x = As[(ai * S_DIM + ka    ) * CH + ac];
            af.y = As[(ai * S_DIM + ka + 1) * CH + ac];
            bf.x = Bs[(bj * S_DIM + ka    ) * CH + bd];
            bf.y = Bs[(bj * S_DIM + ka + 1) * CH + bd];
            acc = __builtin_amdgcn_wmma_f32_16x16x4_f32(
                false, af, false, bf, (short)0, acc, false, false);
        }
        // C/D layout: vgpr v, lane l -> M = v + 8*(l>>4), N = l&15
        #pragma unroll
        for (int v = 0; v < 8; ++v)
            G[(mT + v + 8 * half) * 128 + nT + l15] = acc[v];
    }
    __syncthreads();

    // ---- Stage B: [16 x 1024] x [1024 x 128]; wave w owns z-tile w*16 ----
    {
        const int zT = wave * 16;
        const int pRow = l15;                    // 0..15 -> (i,j) within tile
        const int pi = pRow >> 2, pj = pRow & 3;
        const int zCol = zT + l15;

        v8f acc = {};
        for (int q = 0; q < CH * CH; q += 4) {
            const int q0 = q + 2 * half;
            const int q1 = q0 + 1;
            v2f af, bf;
            af.x = G[(pi * 32 + (q0 >> 5)) * 128 + pj * 32 + (q0 & 31)];
            af.y = G[(pi * 32 + (q1 >> 5)) * 128 + pj * 32 + (q1 & 31)];
            bf.x = w_out[q0 * CZ + zCol];
            bf.y = w_out[q1 * CZ + zCol];
            acc = __builtin_amdgcn_wmma_f32_16x16x4_f32(
                false, af, false, bf, (short)0, acc, false, false);
        }
        const float bo = b_out[zT + l15];
        #pragma unroll
        for (int v = 0; v < 8; ++v) {
            const int p = v + 8 * half;
            const int i = iBase + (p >> 2);
            const int j = jBase + (p & 3);
            out[((size_t)i * N_DIM + j) * CZ + zT + l15] =
                (acc[v] + bo) / normS[p];
        }
    }
}

// ---------------------------------------------------------------------------
extern "C" void kernel_launch(void* const* d_in, const int* in_sizes, int n_in,
                              void* d_out, int out_size, void* d_ws, size_t ws_size,
                              hipStream_t stream) {
    const float* m     = (const float*)d_in[0];
    const float* mask  = (const float*)d_in[1];
    const float* gamma = (const float*)d_in[2];
    const float* beta  = (const float*)d_in[3];
    const float* w1    = (const float*)d_in[4];
    const float* b1    = (const float*)d_in[5];
    const float* w2    = (const float*)d_in[6];
    const float* b2    = (const float*)d_in[7];
    const float* w_out = (const float*)d_in[8];
    const float* b_out = (const float*)d_in[9];
    float* out = (float*)d_out;

    const size_t abElems = (size_t)N_DIM * S_DIM * CH;   // 1.5M floats each
    float* Ag = (float*)d_ws;
    float* Bg = Ag + abElems;

    // Kernel 1: LN + dual projection, transposed outputs.
    opm_ln_proj_kernel<<<S_DIM * N_DIM, 256, 0, stream>>>(
        m, mask, gamma, beta, w1, b1, w2, b2, Ag, Bg);

    // Kernel 2: fused outer-product + w_out projection (WMMA f32).
    const size_t ldsBytes = (2 * TI * S_DIM * CH + 128 * 128) * sizeof(float); // 192 KB
    dim3 grid(N_DIM / TJ, N_DIM / TI);   // 96 x 96
    opm_fused_wmma_kernel<<<grid, 256, ldsBytes, stream>>>(
        Ag, Bg, mask, w_out, b_out, out);
}